// Model_30906584662567
// MI455X (gfx1250) — compile-verified
//
#include <hip/hip_runtime.h>

#define NN 100000
#define EE 2000000
#define DD 256
#define CC 16
#define RR 4

typedef __attribute__((ext_vector_type(2))) float v2f;
typedef __attribute__((ext_vector_type(8))) float v8f;

// ---------------------------------------------------------------------------
// Kernel 0: zero the accumulator region of the workspace (sums + counts).
// ---------------------------------------------------------------------------
__global__ void zero_kernel(float* __restrict__ p, long long n) {
    long long i = (long long)blockIdx.x * blockDim.x + threadIdx.x;
    long long stride = (long long)gridDim.x * blockDim.x;
    for (; i < n; i += stride) p[i] = 0.0f;
}

// ---------------------------------------------------------------------------
// Kernel 1: wh[n, r*16+c] = sum_k feat[n,k] * W[r,k,c] + b[r,c]
// One 128-thread block = 4 waves; wave w computes the 16x16 tile for etype w
// over 16 node rows, using V_WMMA_F32_16X16X4_F32 (K stepped by 4, 64 steps).
// N = 100000 = 6250 * 16 exactly, so every lane is always active (EXEC all-1s
// as required by WMMA).
// ---------------------------------------------------------------------------
__global__ __launch_bounds__(128) void gemm_wmma_kernel(
    const float* __restrict__ feat,   // (N, D)
    const float* __restrict__ W,      // (R, D, C)
    const float* __restrict__ bias,   // (R, C)
    float* __restrict__ wh)           // (N, R*C)
{
    const int wave = threadIdx.x >> 5;       // 0..3 -> etype / column tile
    const int lane = threadIdx.x & 31;
    const int col  = lane & 15;              // N-index (and A row index)
    const int half = lane >> 4;              // selects K pair 0/1 vs 2/3

    const int row0 = blockIdx.x * 16;        // node row base for this block
    const int arow = row0 + col;             // A-matrix row held by this lane

    // A: a.x = feat[arow][k + 2*half], a.y = feat[arow][k + 2*half + 1]
    const float* aptr = feat + (size_t)arow * DD + 2 * half;
    // B: b.x = W[r][k + 2*half][col],  b.y = W[r][k + 2*half + 1][col]
    const float* bptr = W + (size_t)wave * (DD * CC) + (2 * half) * CC + col;

    v8f acc = {};
    #pragma unroll 8
    for (int k = 0; k < DD; k += 4) {
        v2f a, b;
        a.x = aptr[0];
        a.y = aptr[1];
        b.x = bptr[0];
        b.y = bptr[CC];
        acc = __builtin_amdgcn_wmma_f32_16x16x4_f32(
            /*neg_a=*/false, a, /*neg_b=*/false, b,
            /*c_mod=*/(short)0, acc, /*reuse_a=*/false, /*reuse_b=*/false);
        aptr += 4;
        bptr += 4 * CC;
    }

    const float bv = bias[wave * CC + col];

    // D layout: VGPR i holds row M = i + 8*half, column N = col.
    float* out = wh + (size_t)row0 * (RR * CC) + wave * CC + col;
    #pragma unroll
    for (int i = 0; i < 8; ++i) {
        const int m = i + 8 * half;
        out[(size_t)m * (RR * CC)] = acc[i] + bv;
    }
}

// ---------------------------------------------------------------------------
// Kernel 2: edge scatter. One thread per (etype, edge).
//   m = wh[src, r*16 : r*16+16] * ew ; atomicAdd into sums[r, dst, :]
//   atomicAdd 1.0 into cnts[r, dst]
// Each thread's gather is a contiguous 64B cacheline; the 16 f32 atomics land
// in one 64B line of the destination row.
// ---------------------------------------------------------------------------
__global__ __launch_bounds__(256) void scatter_kernel(
    const float* __restrict__ wh,      // (N, R*C)
    const int*   __restrict__ src,     // (R, E) flat
    const int*   __restrict__ dst,     // (R, E) flat
    const float* __restrict__ ew,      // (R, E) flat
    float* __restrict__ sums,          // (R, N, C)
    float* __restrict__ cnts)          // (R, N)
{
    const long long total = (long long)RR * EE;
    long long i = (long long)blockIdx.x * blockDim.x + threadIdx.x;
    if (i >= total) return;

    const int r = (int)(i / EE);
    const int s = src[i];
    const int d = dst[i];
    const float w = ew[i];

    const float4* whp = (const float4*)(wh + (size_t)s * (RR * CC) + r * CC);
    float4 v0 = whp[0];
    float4 v1 = whp[1];
    float4 v2 = whp[2];
    float4 v3 = whp[3];

    float* sp = sums + ((size_t)r * NN + d) * CC;
    atomicAdd(sp + 0,  v0.x * w);
    atomicAdd(sp + 1,  v0.y * w);
    atomicAdd(sp + 2,  v0.z * w);
    atomicAdd(sp + 3,  v0.w * w);
    atomicAdd(sp + 4,  v1.x * w);
    atomicAdd(sp + 5,  v1.y * w);
    atomicAdd(sp + 6,  v1.z * w);
    atomicAdd(sp + 7,  v1.w * w);
    atomicAdd(sp + 8,  v2.x * w);
    atomicAdd(sp + 9,  v2.y * w);
    atomicAdd(sp + 10, v2.z * w);
    atomicAdd(sp + 11, v2.w * w);
    atomicAdd(sp + 12, v3.x * w);
    atomicAdd(sp + 13, v3.y * w);
    atomicAdd(sp + 14, v3.z * w);
    atomicAdd(sp + 15, v3.w * w);

    atomicAdd(cnts + (size_t)r * NN + d, 1.0f);
}

// ---------------------------------------------------------------------------
// Kernel 3: finalize. out[n,c] = sum_r (cnt_r[n] > 0 ? sums_r[n,c]/cnt_r[n] : 0)
// ---------------------------------------------------------------------------
__global__ __launch_bounds__(256) void finalize_kernel(
    const float* __restrict__ sums,    // (R, N, C)
    const float* __restrict__ cnts,    // (R, N)
    float* __restrict__ out)           // (N, C)
{
    const long long total = (long long)NN * CC;
    long long i = (long long)blockIdx.x * blockDim.x + threadIdx.x;
    if (i >= total) return;
    const int n = (int)(i >> 4);
    const int c = (int)(i & 15);

    float acc = 0.0f;
    #pragma unroll
    for (int r = 0; r < RR; ++r) {
        const float cnt = cnts[(size_t)r * NN + n];
        if (cnt > 0.0f) {
            acc += sums[((size_t)r * NN + n) * CC + c] / cnt;
        }
    }
    out[i] = acc;
}

// ---------------------------------------------------------------------------
// Host-side launcher.
// Workspace layout (floats):
//   [0,              N*64)             wh     (N x R*C)
//   [N*64,           2*N*64)           sums   (R x N x C)   <- zeroed
//   [2*N*64,         2*N*64 + R*N)     cnts   (R x N)       <- zeroed
// Total ~52.8 MB.
// ---------------------------------------------------------------------------
extern "C" void kernel_launch(void* const* d_in, const int* in_sizes, int n_in,
                              void* d_out, int out_size, void* d_ws, size_t ws_size,
                              hipStream_t stream) {
    const float* feat = (const float*)d_in[0];   // (N, D)
    const int*   srci = (const int*)  d_in[1];   // (R, E)
    const int*   dsti = (const int*)  d_in[2];   // (R, E)
    const float* ew   = (const float*)d_in[3];   // (R, E, 1)
    const float* W    = (const float*)d_in[4];   // (R, D, C)
    const float* bias = (const float*)d_in[5];   // (R, C)
    float* out = (float*)d_out;                  // (N, C)

    float* ws   = (float*)d_ws;
    float* wh   = ws;                                  // N * 64
    float* sums = ws + (size_t)NN * (RR * CC);         // N * 64
    float* cnts = sums + (size_t)NN * (RR * CC);       // R * N

    // 1) zero accumulators (sums + cnts)
    {
        const long long nz = (long long)NN * (RR * CC) + (long long)RR * NN;
        const int blocks = 4096;
        zero_kernel<<<blocks, 256, 0, stream>>>(sums, nz);
    }

    // 2) fused per-etype linear via f32 WMMA
    {
        dim3 grid(NN / 16);     // 6250 blocks, each = 16 node rows x 4 etypes
        gemm_wmma_kernel<<<grid, 128, 0, stream>>>(feat, W, bias, wh);
    }

    // 3) edge scatter with atomics
    {
        const long long total = (long long)RR * EE;          // 8M
        const int blocks = (int)((total + 255) / 256);       // 31250
        scatter_kernel<<<blocks, 256, 0, stream>>>(wh, srci, dsti, ew, sums, cnts);
    }

    // 4) finalize mean per etype + sum across etypes
    {
        const long long total = (long long)NN * CC;          // 1.6M
        const int blocks = (int)((total + 255) / 256);
        finalize_kernel<<<blocks, 256, 0, stream>>>(sums, cnts, out);
    }
}